// BatchedDifferentiableDynamicBicycleModel_13666585936349
// MI455X (gfx1250) — compile-verified
//
#include <hip/hip_runtime.h>

// ---------------------------------------------------------------------------
// Batched dynamic bicycle model, forward Euler @ 0.01 s internal step.
// Compute-bound elementwise ODE: one thread per vehicle, wave32, hardware
// transcendentals (v_tanh_f32 is new on gfx1250). VALU-issue bound, so the
// inner loop is minimized to ~25 VALU + 5 TRANS ops per substep.
// ---------------------------------------------------------------------------

namespace bike {

// ---- model constants (doubles, folded to f32 at compile time) -------------
constexpr double M_V   = 1500.0;
constexpr double IZ    = 2250.0;
constexpr double LF    = 1.2;
constexpr double LR    = 1.6;
constexpr double CF    = 80000.0;
constexpr double CR    = 80000.0;
constexpr double TAU_A = 0.1;
constexpr double TAU_D = 0.1;
constexpr double PI_D  = 3.14159265358979323846;
constexpr double MAX_STEER_D = 30.0 * PI_D / 180.0;     // 0.5235987755982988
constexpr double MAX_ACC_D   = 3.0;
constexpr double MIN_ACC_D   = -6.0;
constexpr double MU = 0.9, GRAV = 9.81;
constexpr double L_TOT = LF + LR;
constexpr double FY_F_MAX_D = MU * M_V * GRAV * (LR / L_TOT);   // 7567.7142857...
constexpr double FY_R_MAX_D = MU * M_V * GRAV * (LF / L_TOT);   // 5675.7857142...
constexpr double H_D = 0.01;                                    // DT_INTERNAL
constexpr double V_EFF_MIN_D = 20.0 / 3.6;

// f32 fused factors
constexpr float F_H         = (float)H_D;
constexpr float F_LF        = (float)LF;
constexpr float F_LR        = (float)LR;
constexpr float F_MAX_STEER = (float)MAX_STEER_D;
constexpr float F_MAX_ACC   = (float)MAX_ACC_D;
constexpr float F_MIN_ACC   = (float)MIN_ACC_D;
constexpr float F_V_EFF_MIN = (float)V_EFF_MIN_D;
constexpr float F_KFN       = (float)(-CF / FY_F_MAX_D);  // tanh arg scale, front
constexpr float F_KRN       = (float)(-CR / FY_R_MAX_D);  // tanh arg scale, rear
constexpr float F_BF        = (float)(FY_F_MAX_D / M_V);  // (Fyf)/M per unit tanh
constexpr float F_BR        = (float)(FY_R_MAX_D / M_V);
constexpr float F_RFH       = (float)(LF * FY_F_MAX_D / IZ * H_D);  // yaw gain * h
constexpr float F_RRH       = (float)(LR * FY_R_MAX_D / IZ * H_D);
constexpr float F_LAG_H     = (float)(H_D / TAU_A);       // 0.1  (== H_D/TAU_D)
constexpr float F_LAG_DECAY = (float)(1.0 - H_D / TAU_A); // 0.9
constexpr float F_1_2PI     = 0.15915494309189535f;       // 0x3E22F983

} // namespace bike

// ---- hardware transcendentals (gfx1250 TRANS unit) ------------------------
__device__ __forceinline__ float hw_tanh(float x) {
#if __has_builtin(__builtin_amdgcn_tanhf)
    return __builtin_amdgcn_tanhf(x);            // v_tanh_f32 (new on gfx1250)
#else
    float r;
    asm("v_tanh_f32 %0, %1" : "=v"(r) : "v"(x)); // guaranteed CDNA5 path
    return r;
#endif
}
__device__ __forceinline__ float hw_rcp(float x)  { return __builtin_amdgcn_rcpf(x); }
__device__ __forceinline__ float hw_med3(float x, float lo, float hi) {
    return __builtin_amdgcn_fmed3f(x, lo, hi);   // v_med3_f32 clamp
}

__global__ __launch_bounds__(256)
void bicycle_dynamics_kernel(const float* __restrict__ state,   // (B, 9)
                             const float* __restrict__ action,  // (B, 2)
                             const int*   __restrict__ dt_ptr,  // scalar int
                             float*       __restrict__ out,     // (B, 9)
                             int B)
{
    using namespace bike;
    const int i = blockIdx.x * blockDim.x + threadIdx.x;
    if (i >= B) return;

    // For integer dt, Python's (dt // 0.01) + remainder split produces exactly
    // 100*dt substeps whose f32 step sizes are all bitwise 0.01f (the dt=1
    // remainder 0.01000000000000000888 rounds to the same f32 as 0.01), so
    // pure integer math replicates it. Uniform scalar -> s_load + s_mul.
    const int n_total = 100 * (*dt_ptr);

    // ---- load per-vehicle state (one-shot; kernel is compute-bound) -------
    const float* sp = state + (size_t)i * 9;
    float x     = sp[0];
    float y     = sp[1];
    float psi   = sp[2];
    float v     = sp[3];
    float a     = sp[4];
    float delta = sp[5];
    float beta  = sp[6];
    float r     = sp[7];
    // sp[8] (old delta_ref) is unused by the model.

    // action: (B,2), 8-byte aligned -> global_load_b64
    const float2 act = *(const float2*)(action + (size_t)i * 2);
    const float a_ref = hw_med3(act.x, F_MIN_ACC, F_MAX_ACC);
    const float d_ref = hw_med3(act.y, -F_MAX_STEER, F_MAX_STEER);

    // hoisted actuator-lag targets: a' = 0.9*a + (0.1*a_ref)
    const float a_ref_h = a_ref * F_LAG_H;
    const float d_ref_h = d_ref * F_LAG_H;

    // ---- integrate: n_total substeps, h = 0.01f each ----------------------
#pragma unroll 2
    for (int s = 0; s < n_total; ++s) {
        // tire slip & normalized lateral forces (all from old state)
        const float v_eff   = fmaxf(v, F_V_EFF_MIN);
        const float inv_v   = hw_rcp(v_eff);                  // v_rcp_f32
        const float rv      = r * inv_v;
        const float bd      = beta - delta;
        const float alpha_f = fmaf(F_LF, rv, bd);
        const float alpha_r = fmaf(-F_LR, rv, beta);
        const float tf = hw_tanh(F_KFN * alpha_f);            // v_tanh_f32
        const float tr = hw_tanh(F_KRN * alpha_r);            // v_tanh_f32

        // beta_dot = (F_yf + F_yr)/(M*v_eff) - r
        const float fsum     = fmaf(F_BF, tf, F_BR * tr);     // (Fyf+Fyr)/M
        const float beta_dot = fmaf(fsum, inv_v, -r);

        // kinematics (old v, psi, beta, r)
        const float psi_eff = psi + beta;
        const float m  = psi_eff * F_1_2PI;
        const float ch = __builtin_amdgcn_cosf(m);            // v_cos_f32
        const float sh = __builtin_amdgcn_sinf(m);            // v_sin_f32
        const float vh = v * F_H;
        x   = fmaf(vh, ch, x);
        y   = fmaf(vh, sh, y);
        psi = fmaf(r, F_H, psi);                              // old r
        v   = fmaxf(fmaf(a, F_H, v), 0.0f);                   // old a

        // first-order actuator lags: one FMA each (targets hoisted)
        a     = fmaf(a, F_LAG_DECAY, a_ref_h);
        delta = hw_med3(fmaf(delta, F_LAG_DECAY, d_ref_h),
                        -F_MAX_STEER, F_MAX_STEER);           // v_med3_f32

        beta = fmaf(beta_dot, F_H, beta);
        // r += h*(LF*Fyf - LR*Fyr)/IZ  (folded into two FMAs)
        r = fmaf(F_RFH, tf, fmaf(-F_RRH, tr, r));
    }

    // ---- store ------------------------------------------------------------
    float* op = out + (size_t)i * 9;
    op[0] = x;
    op[1] = y;
    op[2] = psi;
    op[3] = v;
    op[4] = a;
    op[5] = delta;
    op[6] = beta;
    op[7] = r;
    op[8] = d_ref;
}

extern "C" void kernel_launch(void* const* d_in, const int* in_sizes, int n_in,
                              void* d_out, int out_size, void* d_ws, size_t ws_size,
                              hipStream_t stream) {
    (void)n_in; (void)out_size; (void)d_ws; (void)ws_size;
    const float* state  = (const float*)d_in[0];   // (B, 9) f32
    const float* action = (const float*)d_in[1];   // (B, 2) f32
    const int*   dt     = (const int*)d_in[2];     // scalar int
    float*       out    = (float*)d_out;           // (B, 9) f32

    const int B = in_sizes[0] / 9;                 // 262144
    const int block = 256;                         // 8 wave32s per workgroup
    const int grid  = (B + block - 1) / block;

    bicycle_dynamics_kernel<<<grid, block, 0, stream>>>(state, action, dt, out, B);
}